// GeometricPerturb_36189394436910
// MI455X (gfx1250) — compile-verified
//
#include <hip/hip_runtime.h>
#include <stdint.h>

// ---------------------------------------------------------------------------
// GeometricPerturb on MI455X (gfx1250)
// Memory-bound (~320 MB traffic -> ~14us @ 23.3 TB/s). One workgroup per
// 32x32 patch; CDNA5 Tensor Data Mover DMAs the patch (img 3x32x32, lbl
// 32x32, pw 32x32) into LDS, 8 wave32 waves gather/interpolate, coalesced
// float4 stores.
// ---------------------------------------------------------------------------

#define NB   8
#define NC   3
#define HH   1024
#define WW   1024
#define BLKP 32
#define NPATCH (NB * (HH / BLKP) * (WW / BLKP))   // 8192
#define TPB  256

typedef __attribute__((ext_vector_type(4))) unsigned int v4u_t;
typedef __attribute__((ext_vector_type(8))) int          v8i_t;
typedef __attribute__((ext_vector_type(4))) int          v4i_t;

#if defined(__HIP_DEVICE_COMPILE__) && __has_builtin(__builtin_amdgcn_tensor_load_to_lds) && __has_builtin(__builtin_amdgcn_s_wait_tensorcnt)
#define USE_TDM 1
#else
#define USE_TDM 0
#endif

__device__ __forceinline__ uint32_t lds_addr_of(const void* p) {
    // Flat LDS addresses are {shared_aperture_hi32, lds_offset}; low 32 bits
    // are the workgroup-relative LDS byte offset the TDM descriptor wants.
    return (uint32_t)(uintptr_t)p;
}

#if USE_TDM
// Build a Tensor DMA Descriptor (ISA cdna5 ch.8) for a 32-bit-element tile
// and issue TENSOR_LOAD_TO_LDS. tile2==0 -> plain 2D tile; tile2>0 -> 3D
// tile where the Z loop advances by stride1 (element units).
// This toolchain's builtin is the 6-arg clang-23 form:
//   (uint32x4 g0, int32x8 g1, int32x4 g2, int32x4 g3, int32x8 zero, i32 cpol)
__device__ __forceinline__ void tdm_tile_load(
    uint32_t lds, const void* gptr,
    uint32_t tile0, uint32_t tile1, uint32_t tile2,
    uint32_t tdim0, uint32_t tdim1, uint32_t tdim2,
    uint64_t stride0, uint64_t stride1)
{
    const uint64_t ga = (uint64_t)(uintptr_t)gptr;
    v4u_t g0;
    g0.x = 1u;                                    // count=1, user-mode load
    g0.y = lds;                                   // lds_addr (bytes)
    g0.z = (uint32_t)ga;                          // global_addr[31:0]
    g0.w = (uint32_t)((ga >> 32) & 0x01FFFFFFu)   // global_addr[56:32]
         | 0x80000000u;                           // type=2 ("image")
    v8i_t g1;
    g1[0] = (int)(2u << 16);                      // data_size=2 -> 4 bytes
    g1[1] = (int)((tdim0 & 0xFFFFu) << 16);       // tensor_dim0[15:0]
    g1[2] = (int)(((tdim0 >> 16) & 0xFFFFu) | ((tdim1 & 0xFFFFu) << 16));
    g1[3] = (int)(((tdim1 >> 16) & 0xFFFFu) | ((tile0 & 0xFFFFu) << 16));
    g1[4] = (int)((tile1 & 0xFFFFu) | ((tile2 & 0xFFFFu) << 16));
    g1[5] = (int)(uint32_t)(stride0 & 0xFFFFFFFFull);               // dim0_stride lo
    g1[6] = (int)(uint32_t)(((stride0 >> 32) & 0xFFFFull)
                            | ((stride1 & 0xFFFFull) << 16));       // hi16 | dim1_stride lo16
    g1[7] = (int)(uint32_t)((stride1 >> 16) & 0xFFFFFFFFull);       // dim1_stride hi32
    v4i_t g2;
    g2[0] = (int)tdim2;   // tensor_dim2
    g2[1] = 0;            // tensor_dim3
    g2[2] = 0;            // tensor_dim2_stride lo
    g2[3] = 0;            // tensor_dim2_stride hi | tile_dim3=0
    v4i_t g3 = (v4i_t){0, 0, 0, 0};
    v8i_t gz = (v8i_t){0, 0, 0, 0, 0, 0, 0, 0};   // zero-filled 5th operand
    __builtin_amdgcn_tensor_load_to_lds(g0, g1, g2, g3, gz, 0);
}
#endif

__global__ void __launch_bounds__(TPB)
geo_perturb_kernel(const float* __restrict__ imgs,
                   const int*   __restrict__ lbls,
                   const float* __restrict__ pwt,
                   const float* __restrict__ theta,
                   float* __restrict__ o_img,
                   float* __restrict__ o_lbl,
                   float* __restrict__ o_pw)
{
    __shared__ float s_img[NC * BLKP * BLKP];   // 12 KB
    __shared__ int   s_lbl[BLKP * BLKP];        //  4 KB
    __shared__ float s_pw [BLKP * BLKP];        //  4 KB

    const int n  = blockIdx.x;                  // patch id, matches patchify order
    const int bi = n >> 10;                     // batch
    const int gy = (n >> 5) & 31;               // H-block
    const int gx = n & 31;                      // W-block
    const int y0 = gy << 5;
    const int x0 = gx << 5;

#if USE_TDM
    if (threadIdx.x == 0) {
        const float* gi = imgs + (((size_t)bi * NC) * HH + (size_t)y0) * WW + x0;
        tdm_tile_load(lds_addr_of(s_img), gi, BLKP, BLKP, NC,
                      WW, HH, NC, (uint64_t)WW, (uint64_t)HH * WW);
        const int* gl = lbls + ((size_t)bi * HH + (size_t)y0) * WW + x0;
        tdm_tile_load(lds_addr_of(s_lbl), gl, BLKP, BLKP, 0,
                      WW, HH, 0, (uint64_t)WW, 0);
        const float* gp = pwt + ((size_t)bi * HH + (size_t)y0) * WW + x0;
        tdm_tile_load(lds_addr_of(s_pw), gp, BLKP, BLKP, 0,
                      WW, HH, 0, (uint64_t)WW, 0);
        __builtin_amdgcn_s_wait_tensorcnt(0);   // TDM done -> LDS valid
    }
#else
    // Cooperative fallback: coalesced float4 loads into LDS.
    for (int j = threadIdx.x; j < NC * 256; j += TPB) {
        const int ch = j >> 8, jj = j & 255;
        const int rr = jj >> 3, cc = (jj & 7) << 2;
        *(float4*)&s_img[ch * (BLKP * BLKP) + rr * BLKP + cc] =
            *(const float4*)(imgs + (((size_t)(bi * NC + ch)) * HH + y0 + rr) * WW + x0 + cc);
    }
    {
        const int jj = threadIdx.x;
        const int rr = jj >> 3, cc = (jj & 7) << 2;
        *(int4*)&s_lbl[rr * BLKP + cc] =
            *(const int4*)(lbls + ((size_t)bi * HH + y0 + rr) * WW + x0 + cc);
        *(float4*)&s_pw[rr * BLKP + cc] =
            *(const float4*)(pwt + ((size_t)bi * HH + y0 + rr) * WW + x0 + cc);
    }
#endif
    __syncthreads();

    // Block-uniform inverse affine (n is uniform -> scalar loads/regs).
    const float* th  = theta + (size_t)n * 6;
    const float a00 = th[0], a01 = th[1], t0 = th[2];
    const float a10 = th[3], a11 = th[4], t1 = th[5];
    const float rdet = 1.0f / (a00 * a11 - a01 * a10);
    const float i00 =  a11 * rdet, i01 = -a01 * rdet;
    const float i10 = -a10 * rdet, i11 =  a00 * rdet;
    const float i02 = (a01 * t1 - a11 * t0) * rdet;
    const float i12 = (a10 * t0 - a00 * t1) * rdet;

    const int t  = threadIdx.x;
    const int r  = t >> 3;          // row 0..31
    const int c0 = (t & 7) << 2;    // col base 0,4,...,28

    float vimg[NC][4];
    float vlbl[4];
    float vpw[4];

#pragma unroll
    for (int k = 0; k < 4; ++k) {
        const int c = c0 + k;
        const float sx = fmaf(i00, (float)c, fmaf(i01, (float)r, i02));
        const float sy = fmaf(i10, (float)c, fmaf(i11, (float)r, i12));

        // ---- bilinear taps (images), zeros padding via weight masking ----
        const float fx0 = floorf(sx), fy0 = floorf(sy);
        const float wx = sx - fx0, wy = sy - fy0;
        const int xi = (int)fx0, yi = (int)fy0;
        const bool bx0 = (xi     >= 0) && (xi     <= BLKP - 1);
        const bool bx1 = (xi + 1 >= 0) && (xi + 1 <= BLKP - 1);
        const bool by0 = (yi     >= 0) && (yi     <= BLKP - 1);
        const bool by1 = (yi + 1 >= 0) && (yi + 1 <= BLKP - 1);
        const int cx0 = min(max(xi,     0), BLKP - 1);
        const int cx1 = min(max(xi + 1, 0), BLKP - 1);
        const int cy0 = min(max(yi,     0), BLKP - 1);
        const int cy1 = min(max(yi + 1, 0), BLKP - 1);
        const float w00 = (bx0 && by0) ? (1.0f - wx) * (1.0f - wy) : 0.0f;
        const float w10 = (bx1 && by0) ? wx * (1.0f - wy)          : 0.0f;
        const float w01 = (bx0 && by1) ? (1.0f - wx) * wy          : 0.0f;
        const float w11 = (bx1 && by1) ? wx * wy                   : 0.0f;
        const int s00 = cy0 * BLKP + cx0;
        const int s10 = cy0 * BLKP + cx1;
        const int s01 = cy1 * BLKP + cx0;
        const int s11 = cy1 * BLKP + cx1;
        const int hidx = r * BLKP + c;

#pragma unroll
        for (int ch = 0; ch < NC; ++ch) {
            const float* sp = s_img + ch * (BLKP * BLKP);
            float acc = sp[s00] * w00;
            acc = fmaf(sp[s10], w10, acc);
            acc = fmaf(sp[s01], w01, acc);
            acc = fmaf(sp[s11], w11, acc);
            vimg[ch][k] = (acc == 0.0f) ? sp[hidx] : acc;   // padding -> original
        }

        // ---- nearest (labels + pixel weights), round-half-even = jnp.round ----
        const int xn = (int)rintf(sx);
        const int yn = (int)rintf(sy);
        const bool inn = (xn >= 0) && (xn <= BLKP - 1) && (yn >= 0) && (yn <= BLKP - 1);
        const int sidx = min(max(yn, 0), BLKP - 1) * BLKP + min(max(xn, 0), BLKP - 1);

        int ls = s_lbl[sidx]; ls = (ls == 0) ? -1 : ls;     // 0 -> -1 sentinel
        const int lw = inn ? ls : 0;                        // zeros padding
        int lh = s_lbl[hidx]; lh = (lh == 0) ? -1 : lh;
        const int lm = (lw == 0) ? lh : lw;                 // padding -> original
        vlbl[k] = (float)((lm == -1) ? 0 : lm);             // -1 sentinel -> 0

        float ps = s_pw[sidx]; ps = (ps == 0.0f) ? -1.0f : ps;
        const float pwv = inn ? ps : 0.0f;
        float ph = s_pw[hidx]; ph = (ph == 0.0f) ? -1.0f : ph;
        const float pm = (pwv == 0.0f) ? ph : pwv;
        vpw[k] = (pm == -1.0f) ? 0.0f : pm;
    }

    // ---- coalesced float4 stores ----
    const size_t row = (size_t)(y0 + r) * WW + (size_t)(x0 + c0);
#pragma unroll
    for (int ch = 0; ch < NC; ++ch) {
        *(float4*)(o_img + (size_t)(bi * NC + ch) * HH * WW + row) =
            make_float4(vimg[ch][0], vimg[ch][1], vimg[ch][2], vimg[ch][3]);
    }
    *(float4*)(o_lbl + (size_t)bi * HH * WW + row) =
        make_float4(vlbl[0], vlbl[1], vlbl[2], vlbl[3]);
    *(float4*)(o_pw  + (size_t)bi * HH * WW + row) =
        make_float4(vpw[0], vpw[1], vpw[2], vpw[3]);
}

extern "C" void kernel_launch(void* const* d_in, const int* in_sizes, int n_in,
                              void* d_out, int out_size, void* d_ws, size_t ws_size,
                              hipStream_t stream) {
    (void)in_sizes; (void)n_in; (void)out_size; (void)d_ws; (void)ws_size;
    const float* imgs  = (const float*)d_in[0];   // (8,3,1024,1024) f32
    const int*   lbls  = (const int*)  d_in[1];   // (8,1,1024,1024) i32
    const float* pwt   = (const float*)d_in[2];   // (8,1024,1024)   f32
    const float* theta = (const float*)d_in[3];   // (8192,2,3)      f32

    float* out   = (float*)d_out;                 // pimg | plbl | ppw, concat flat
    float* o_img = out;
    float* o_lbl = out + (size_t)NB * NC * HH * WW;
    float* o_pw  = o_lbl + (size_t)NB * HH * WW;

    hipLaunchKernelGGL(geo_perturb_kernel, dim3(NPATCH), dim3(TPB), 0, stream,
                       imgs, lbls, pwt, theta, o_img, o_lbl, o_pw);
}